// SupervisedGraphSage_16535624090308
// MI455X (gfx1250) — compile-verified
//
#include <hip/hip_runtime.h>

typedef __attribute__((ext_vector_type(2))) float v2f;
typedef __attribute__((ext_vector_type(8))) float v8f;

#define GS_ALPHA 0.2f

static constexpr int GN = 100000;  // nodes
static constexpr int GF = 512;     // input features
static constexpr int GD = 256;     // hidden dim
static constexpr int GC = 40;      // classes
static constexpr int GB = 16384;   // batch
static constexpr int GS1 = 5;      // layer-1 neighbor samples
static constexpr int GS2 = 10;     // layer-2 neighbor samples

// ---------------------------------------------------------------------------
// agg1[i,:] = (x[i,:] + sum_j x[neigh1[i,j],:]) / (S1+1)   for all N nodes
// One block per node row, 128 threads x float4 = 512 floats.
// ---------------------------------------------------------------------------
__global__ void __launch_bounds__(128) agg1_kernel(const float* __restrict__ x,
                                                   const int* __restrict__ neigh1,
                                                   float* __restrict__ agg1) {
  const int i = blockIdx.x;
  const int t = threadIdx.x;                  // 0..127
  const size_t base = (size_t)i * GF + 4 * t;
  float4 s = *(const float4*)(x + base);
#pragma unroll
  for (int j = 0; j < GS1; ++j) {
    const int nj = neigh1[i * GS1 + j];       // uniform per block -> scalar load
    const float4 v = *(const float4*)(x + (size_t)nj * GF + 4 * t);
    s.x += v.x; s.y += v.y; s.z += v.z; s.w += v.w;
  }
  const float inv = 1.0f / (float)(GS1 + 1);
  s.x *= inv; s.y *= inv; s.z *= inv; s.w *= inv;
  *(float4*)(agg1 + base) = s;
}

// ---------------------------------------------------------------------------
// agg2[i,:] = (h1[nodes[i],:] + sum_j h1[neigh2[i,j],:]) / (S2+1)  for batch
// One block per batch row, 64 threads x float4 = 256 floats.
// ---------------------------------------------------------------------------
__global__ void __launch_bounds__(64) agg2_kernel(const float* __restrict__ h1,
                                                  const int* __restrict__ nodes,
                                                  const int* __restrict__ neigh2,
                                                  float* __restrict__ agg2) {
  const int i = blockIdx.x;
  const int t = threadIdx.x;                  // 0..63
  const int self = nodes[i];
  float4 s = *(const float4*)(h1 + (size_t)self * GD + 4 * t);
#pragma unroll
  for (int j = 0; j < GS2; ++j) {
    const int nj = neigh2[i * GS2 + j];
    const float4 v = *(const float4*)(h1 + (size_t)nj * GD + 4 * t);
    s.x += v.x; s.y += v.y; s.z += v.z; s.w += v.w;
  }
  const float inv = 1.0f / (float)(GS2 + 1);
  s.x *= inv; s.y *= inv; s.z *= inv; s.w *= inv;
  *(float4*)(agg2 + (size_t)i * GD + 4 * t) = s;
}

// ---------------------------------------------------------------------------
// C[M,N] = act(A[M,K] @ W[N,K]^T), row-major everywhere (NT GEMM).
// One wave (32 threads) computes a 32x64 output tile: 8 v8f accumulators,
// K-loop in steps of 4 feeding V_WMMA_F32_16X16X4_F32. Per k-step: 2 A
// fragments + 4 B fragments (6 b64 loads) feed 8 WMMAs, and the loop is
// software-pipelined one chunk ahead so the loads for chunk k+1 are in
// flight while chunk k's WMMAs execute (avoids s_wait_loadcnt 0 stalls).
//
// Fragment layout (ISA 7.12.2, 32-bit A 16x4 and its mirror for B 4x16):
//   A: lane holds A[m, k0+kk .. k0+kk+1], m = lane&15, kk = (lane>>4)*2
//   B: lane holds W[n, k0+kk .. k0+kk+1], n = subtile_origin + (lane&15)
//   C: VGPR r -> row (lane<16 ? r : r+8), col = lane&15
// M must be a multiple of 32 and K of 4 (true at all call sites). GUARD=1
// (classifier, N=40) clamps B row pointers in-bounds and zero-masks padded
// columns so EXEC stays all-1s at the WMMA as the ISA requires; GUARD=0
// compiles the masks away entirely.
// ---------------------------------------------------------------------------
template <int RELU, int GUARD>
__global__ void __launch_bounds__(32) gemm_nt_wmma_f32(const float* __restrict__ A,
                                                       const float* __restrict__ W,
                                                       float* __restrict__ C,
                                                       int M, int N, int K) {
  const int tn = blockIdx.x << 6;             // 64-wide N tile origin
  const int tm = blockIdx.y << 5;             // 32-tall M tile origin
  const int lane = threadIdx.x;               // 0..31 (wave32)
  const int r = lane & 15;
  const int kk = (lane >> 4) << 1;

  const float* __restrict__ Arow0 = A + (size_t)(tm + r) * K + kk;
  const float* __restrict__ Arow1 = Arow0 + (size_t)16 * K;

  const float* Wrow[4];
  float bmask[4];
#pragma unroll
  for (int t = 0; t < 4; ++t) {
    const int bn = tn + 16 * t + r;
    const int bnc = GUARD ? ((bn < N) ? bn : (N - 1)) : bn;  // stay in-bounds
    bmask[t] = (bn < N) ? 1.0f : 0.0f;
    Wrow[t] = W + (size_t)bnc * K + kk;
  }

  v8f acc0[4] = {};
  v8f acc1[4] = {};

  // ---- software pipeline: prologue loads chunk 0 -------------------------
  v2f a0c = *(const v2f*)(Arow0);
  v2f a1c = *(const v2f*)(Arow1);
  v2f bc[4];
#pragma unroll
  for (int t = 0; t < 4; ++t) {
    bc[t] = *(const v2f*)(Wrow[t]);
    if (GUARD) { bc[t].x *= bmask[t]; bc[t].y *= bmask[t]; }
  }

  for (int k = 4; k < K; k += 4) {
    // issue next chunk's loads first so they overlap the WMMAs below
    v2f a0n = *(const v2f*)(Arow0 + k);
    v2f a1n = *(const v2f*)(Arow1 + k);
    v2f bn[4];
#pragma unroll
    for (int t = 0; t < 4; ++t) bn[t] = *(const v2f*)(Wrow[t] + k);

#pragma unroll
    for (int t = 0; t < 4; ++t)
      acc0[t] = __builtin_amdgcn_wmma_f32_16x16x4_f32(false, a0c, false, bc[t],
                                                      (short)0, acc0[t], false, false);
#pragma unroll
    for (int t = 0; t < 4; ++t)
      acc1[t] = __builtin_amdgcn_wmma_f32_16x16x4_f32(false, a1c, false, bc[t],
                                                      (short)0, acc1[t], false, false);

    a0c = a0n;
    a1c = a1n;
#pragma unroll
    for (int t = 0; t < 4; ++t) {
      bc[t] = bn[t];
      if (GUARD) { bc[t].x *= bmask[t]; bc[t].y *= bmask[t]; }
    }
  }

  // ---- epilogue: last chunk's WMMAs --------------------------------------
#pragma unroll
  for (int t = 0; t < 4; ++t)
    acc0[t] = __builtin_amdgcn_wmma_f32_16x16x4_f32(false, a0c, false, bc[t],
                                                    (short)0, acc0[t], false, false);
#pragma unroll
  for (int t = 0; t < 4; ++t)
    acc1[t] = __builtin_amdgcn_wmma_f32_16x16x4_f32(false, a1c, false, bc[t],
                                                    (short)0, acc1[t], false, false);

  // ---- store 32x64 tile with fused activation ----------------------------
  const int rbase0 = tm + ((lane >> 4) << 3);
  const int rbase1 = rbase0 + 16;
#pragma unroll
  for (int t = 0; t < 4; ++t) {
    const int col = tn + 16 * t + r;
    if (!GUARD || col < N) {
#pragma unroll
      for (int rr = 0; rr < 8; ++rr) {
        float v0 = acc0[t][rr];
        float v1 = acc1[t][rr];
        if (RELU) {
          v0 = (v0 >= 0.0f) ? v0 : GS_ALPHA * v0;
          v1 = (v1 >= 0.0f) ? v1 : GS_ALPHA * v1;
        }
        C[(size_t)(rbase0 + rr) * N + col] = v0;
        C[(size_t)(rbase1 + rr) * N + col] = v1;
      }
    }
  }
}

// ---------------------------------------------------------------------------
// Host-side orchestration. Inputs (setup_inputs order):
//   0:x[N,F] 1:W1[D,F] 2:W2[D,D] 3:Wc[C,D] 4:nodes[B] 5:neigh1[N,S1] 6:neigh2[B,S2]
// Output: scores[B,C] float32.
// Workspace: agg1 | h1 | agg2 | h2  (~341 MB)
// ---------------------------------------------------------------------------
extern "C" void kernel_launch(void* const* d_in, const int* in_sizes, int n_in,
                              void* d_out, int out_size, void* d_ws, size_t ws_size,
                              hipStream_t stream) {
  (void)in_sizes; (void)n_in; (void)out_size; (void)ws_size;

  const float* x      = (const float*)d_in[0];
  const float* W1     = (const float*)d_in[1];
  const float* W2     = (const float*)d_in[2];
  const float* Wc     = (const float*)d_in[3];
  const int*   nodes  = (const int*)d_in[4];
  const int*   neigh1 = (const int*)d_in[5];
  const int*   neigh2 = (const int*)d_in[6];
  float*       out    = (float*)d_out;

  float* agg1 = (float*)d_ws;
  float* h1   = agg1 + (size_t)GN * GF;
  float* agg2 = h1 + (size_t)GN * GD;
  float* h2   = agg2 + (size_t)GB * GD;

  // Layer 1: gather+mean over all N nodes, then h1 = leaky_relu(agg1 @ W1^T)
  agg1_kernel<<<GN, 128, 0, stream>>>(x, neigh1, agg1);
  gemm_nt_wmma_f32<1, 0><<<dim3(GD / 64, GN / 32), 32, 0, stream>>>(agg1, W1, h1,
                                                                    GN, GD, GF);
  // Layer 2: gather+mean on batch, then h2 = leaky_relu(agg2 @ W2^T)
  agg2_kernel<<<GB, 64, 0, stream>>>(h1, nodes, neigh2, agg2);
  gemm_nt_wmma_f32<1, 0><<<dim3(GD / 64, GB / 32), 32, 0, stream>>>(agg2, W2, h2,
                                                                    GB, GD, GD);
  // Classifier: scores = h2 @ Wc^T (no activation), N=40 inside one 64-wide tile
  gemm_nt_wmma_f32<0, 1><<<dim3((GC + 63) / 64, GB / 32), 32, 0, stream>>>(h2, Wc, out,
                                                                           GB, GC, GD);
}